// TNN_10385230922350
// MI455X (gfx1250) — compile-verified
//
#include <hip/hip_runtime.h>
#include <cstdint>

typedef __attribute__((ext_vector_type(2))) float        v2f;
typedef __attribute__((ext_vector_type(8))) float        v8f;
typedef __attribute__((ext_vector_type(4))) unsigned int v4u;
typedef __attribute__((ext_vector_type(4))) int          v4i;
typedef __attribute__((ext_vector_type(8))) int          v8i;

#define N_I      64
#define DDIM     4096
#define BATCH    8192
#define LDSPITCH 80               // 64 data dwords + 16 pad dwords (TDM-inserted)
#define LDSCOMP  (N_I * LDSPITCH) // floats per component in LDS (5120)

// ---------------------------------------------------------------------------
// Kernel 1: build A = Psi^T = exp(1j*phase)^T as separate cos/sin tables.
// A_re[i*64+j] = cos(phase[j*64+i]); A_im[i*64+j] = sin(phase[j*64+i]).
// ---------------------------------------------------------------------------
__global__ void psi_build_kernel(const float* __restrict__ phase,
                                 float* __restrict__ Are,
                                 float* __restrict__ Aim) {
  int idx = blockIdx.x * blockDim.x + threadIdx.x;   // 0..4095
  int i = idx >> 6;
  int j = idx & 63;
  float p = phase[j * 64 + i];
  float s, c;
  __sincosf(p, &s, &c);
  Are[idx] = c;
  Aim[idx] = s;
}

// ---------------------------------------------------------------------------
// fp32 WMMA helper: D = A(16x4) x B(4x16) + C(16x16)
// ---------------------------------------------------------------------------
__device__ __forceinline__ v8f wmma4(v2f a, v2f b, v8f c) {
  return __builtin_amdgcn_wmma_f32_16x16x4_f32(false, a, false, b,
                                               (short)0, c, false, false);
}

// ---------------------------------------------------------------------------
// TDM: DMA 4096 x f32 (16 KB) from global into LDS, inserting 16 pad dwords
// every 64 dwords (row pitch 64 -> 80) so the four WMMA rows per DS op land
// in disjoint LDS bank windows ({0,16,32,48}+col). D# per CDNA5 ISA.
// ---------------------------------------------------------------------------
__device__ __forceinline__ void tdm_load_row_padded(unsigned lds_off,
                                                    const void* gptr) {
  uint64_t ga = (uint64_t)(uintptr_t)gptr;
  v4u g0;
  g0.x = 1u;                                               // count=1, user desc
  g0.y = lds_off;                                          // lds_addr (bytes)
  g0.z = (unsigned)ga;                                     // global_addr[31:0]
  g0.w = ((unsigned)(ga >> 32) & 0x1FFFFFFu) | (2u << 30); // addr[56:32]|type=2
  const int w0 = (2 << 16)      // data_size = 4 bytes
               | (1 << 20)      // pad_enable
               | (5 << 22)      // pad_interval: every 64 dwords
               | (15 << 25);    // pad_amount: 16 dwords
  v8i g1 = {
    w0,
    4096 << 16,         // tensor_dim0 = 4096 (bits 79:48, low half)
    1 << 16,            // tensor_dim1 = 1
    4096 << 16,         // tile_dim0 = 4096
    1,                  // tile_dim1 = 1, tile_dim2 = 0
    4096,               // tensor_dim0_stride = 4096
    0, 0                // tensor_dim1_stride = 0
  };
  v4i g2 = {0, 0, 0, 0};
  v4i g3 = {0, 0, 0, 0};
#if __clang_major__ >= 23
  v8i g4 = {0, 0, 0, 0, 0, 0, 0, 0};
  __builtin_amdgcn_tensor_load_to_lds(g0, g1, g2, g3, g4, 0);
#else
  __builtin_amdgcn_tensor_load_to_lds(g0, g1, g2, g3, 0);
#endif
}

// ---------------------------------------------------------------------------
// Main kernel: one workgroup (8 waves) per batch.
//   out[b,i,k] = sum_j A[i,j] * H[b,j,k]   (complex, M=K=64, per-batch N=64)
// 3-multiply complex GEMM: re = RR - II ; im = SS - RR - II
// Wave w: i-tile = w&3 (16 rows), k-tiles = {2*(w>>2), 2*(w>>2)+1}.
// ---------------------------------------------------------------------------
__global__ void __launch_bounds__(256)
tnn_wmma_kernel(const float* __restrict__ h_re,
                const float* __restrict__ h_im,
                const float* __restrict__ Are,
                const float* __restrict__ Aim,
                float* __restrict__ out) {
  __shared__ float sh[2 * LDSCOMP];           // [H_re 20KB][H_im 20KB], padded
  const int b    = blockIdx.x;
  const int tid  = threadIdx.x;
  const int w    = tid >> 5;
  const int lane = tid & 31;

  // --- Stage this batch's H_re / H_im into LDS via the Tensor Data Mover.
  //     Waves 0 and 1 each issue one TDM op (EXEC-independent, per-wave).
  if (w < 2) {
    const float* src = (w == 0) ? (h_re + (size_t)b * DDIM)
                                : (h_im + (size_t)b * DDIM);
    unsigned lds_off = (unsigned)(uintptr_t)(&sh[w * LDSCOMP]);
    tdm_load_row_padded(lds_off, src);
  }

  // --- While the TDM is in flight, pull A fragments (L2-resident) into VGPRs.
  // A-matrix 16x4 f32 layout: lanes 0-15 -> K={0,1}, lanes 16-31 -> K={2,3}.
  const int itile = w & 3;
  const int half  = lane >> 4;         // 0 or 1
  const int lrow  = lane & 15;
  const int ar    = itile * 16 + lrow; // A row for this lane
  const int acb   = half * 2;          // A col base within a K-step
  v2f are[16], aim[16], asum[16];
#pragma unroll
  for (int ks = 0; ks < 16; ++ks) {
    const int c = ks * 4 + acb;
    are[ks]  = *(const v2f*)(Are + ar * 64 + c);
    aim[ks]  = *(const v2f*)(Aim + ar * 64 + c);
    asum[ks] = are[ks] + aim[ks];
  }

  __builtin_amdgcn_s_wait_tensorcnt(0);
  __syncthreads();

  const float* shR  = sh;
  const float* shI  = sh + LDSCOMP;
  float*       outp = out + 2 * ((size_t)b * DDIM);  // interleaved complex64
  const int ntbase  = (w >> 2) * 2;

#pragma unroll
  for (int t = 0; t < 2; ++t) {
    const int nt  = ntbase + t;
    const int col = nt * 16 + lrow;                  // output column (k)
    v8f acc_rr = {}, acc_ii = {}, acc_ss = {};
#pragma unroll
    for (int ks = 0; ks < 16; ++ks) {
      // B-matrix 4x16 f32: lanes 0-15 rows {j0,j0+1}, lanes 16-31 {j0+2,j0+3}
      const int r0 = ks * 4 + acb;
      v2f bre, bim;
      bre.x = shR[r0 * LDSPITCH + col];
      bre.y = shR[(r0 + 1) * LDSPITCH + col];
      bim.x = shI[r0 * LDSPITCH + col];
      bim.y = shI[(r0 + 1) * LDSPITCH + col];
      v2f bsm = bre + bim;
      acc_rr = wmma4(are[ks],  bre, acc_rr);   // Re(A)*Re(H)
      acc_ii = wmma4(aim[ks],  bim, acc_ii);   // Im(A)*Im(H)
      acc_ss = wmma4(asum[ks], bsm, acc_ss);   // (Re+Im)(Re+Im)
    }
    // C/D 16x16 layout: VGPR v, lanes 0-15 -> M=v, lanes 16-31 -> M=v+8
#pragma unroll
    for (int v = 0; v < 8; ++v) {
      const int row = itile * 16 + v + 8 * half;
      const float re = acc_rr[v] - acc_ii[v];
      const float im = acc_ss[v] - acc_rr[v] - acc_ii[v];
      v2f val = {re, im};
      // Streaming output: non-temporal store, keep L2 for reused data.
      __builtin_nontemporal_store(val, (v2f*)(outp + 2 * (row * 64 + col)));
    }
  }
}

// ---------------------------------------------------------------------------
extern "C" void kernel_launch(void* const* d_in, const int* in_sizes, int n_in,
                              void* d_out, int out_size, void* d_ws, size_t ws_size,
                              hipStream_t stream) {
  const float* phase = (const float*)d_in[0];   // (64,64) f32
  const float* h_re  = (const float*)d_in[1];   // (8192,4096) f32
  const float* h_im  = (const float*)d_in[2];   // (8192,4096) f32
  float* Are = (float*)d_ws;                    // 4096 f32
  float* Aim = Are + 4096;                      // 4096 f32

  psi_build_kernel<<<16, 256, 0, stream>>>(phase, Are, Aim);
  tnn_wmma_kernel<<<BATCH, 256, 0, stream>>>(h_re, h_im, Are, Aim, (float*)d_out);
}